// CoNaTEncoder_66108136620627
// MI455X (gfx1250) — compile-verified
//
#include <hip/hip_runtime.h>
#include <math.h>

typedef __attribute__((ext_vector_type(2))) float v2f;
typedef __attribute__((ext_vector_type(8))) float v8f;

#if defined(__AMDGCN__)
#if __has_builtin(__builtin_amdgcn_global_load_async_to_lds_b32) && \
    __has_builtin(__builtin_amdgcn_s_wait_asynccnt)
#define USE_ASYNC_LDS 1
#endif
#endif

#ifdef USE_ASYNC_LDS
typedef __attribute__((address_space(3))) int lds_i;
typedef __attribute__((address_space(1))) int glb_i;
#endif

__device__ __forceinline__ float gelu_exact(float x) {
    return 0.5f * x * (1.0f + erff(x * 0.70710678118654752440f));
}

// ---------------- conv 3x3x3 SAME, NCDHW, zero padding ----------------
// fuse_res == 0: out = conv + bias
// fuse_res == 1: out = gelu(conv + bias) + res     (conv-block tail, saves a pass)
__global__ void conv3d_kernel(const float* __restrict__ in, const float* __restrict__ w,
                              const float* __restrict__ bias, const float* __restrict__ res,
                              float* __restrict__ out, int Cin, int Cout, int D, int fuse_res) {
    const int Nv = D * D * D;
    int idx = blockIdx.x * blockDim.x + threadIdx.x;
    if (idx >= Nv * Cout) return;
    int co = idx / Nv;
    int v  = idx - co * Nv;
    int z  = v % D;
    int yy = (v / D) % D;
    int xx = v / (D * D);
    float acc = bias[co];
    const float* wc = w + (size_t)co * Cin * 27;
    for (int ci = 0; ci < Cin; ++ci) {
        const float* ip = in + (size_t)ci * Nv;
        if (ci + 1 < Cin) __builtin_prefetch(in + (size_t)(ci + 1) * Nv + v, 0, 3);
        const float* wp = wc + ci * 27;
        int n = 0;
        for (int kd = -1; kd <= 1; ++kd)
            for (int kh = -1; kh <= 1; ++kh)
                for (int kw = -1; kw <= 1; ++kw, ++n) {
                    int d2 = xx + kd, h2 = yy + kh, w2 = z + kw;
                    if (d2 >= 0 && d2 < D && h2 >= 0 && h2 < D && w2 >= 0 && w2 < D)
                        acc += wp[n] * ip[((size_t)d2 * D + h2) * D + w2];
                }
    }
    if (fuse_res)
        out[(size_t)co * Nv + v] = gelu_exact(acc) + res[(size_t)co * Nv + v];
    else
        out[(size_t)co * Nv + v] = acc;
}

// ---------------- instance norm: per-channel mean / rstd ----------------
__global__ void in_stats_kernel(const float* __restrict__ in, float* __restrict__ stats, int Nv) {
    int c = blockIdx.x;
    const float* p = in + (size_t)c * Nv;
    float s = 0.f, s2 = 0.f;
    for (int i = threadIdx.x; i < Nv; i += blockDim.x) {
        float x = p[i];
        s += x; s2 += x * x;
    }
    __shared__ float sh[256], sh2[256];
    sh[threadIdx.x] = s; sh2[threadIdx.x] = s2;
    __syncthreads();
    for (int o = blockDim.x >> 1; o > 0; o >>= 1) {
        if ((int)threadIdx.x < o) {
            sh[threadIdx.x]  += sh[threadIdx.x + o];
            sh2[threadIdx.x] += sh2[threadIdx.x + o];
        }
        __syncthreads();
    }
    if (threadIdx.x == 0) {
        float m   = sh[0] / (float)Nv;
        float var = sh2[0] / (float)Nv - m * m;
        stats[2 * c]     = m;
        stats[2 * c + 1] = rsqrtf(var + 1e-5f);
    }
}

__global__ void norm_gelu_kernel(const float* __restrict__ in, const float* __restrict__ stats,
                                 float* __restrict__ out, int Nv, int total) {
    int i = blockIdx.x * blockDim.x + threadIdx.x;
    if (i >= total) return;
    int c = i / Nv;
    float m = stats[2 * c], r = stats[2 * c + 1];
    out[i] = gelu_exact((in[i] - m) * r);
}

// ---------------- 2x2x2 average pool ----------------
__global__ void avgpool2_kernel(const float* __restrict__ in, float* __restrict__ out,
                                int C, int D) {
    int Do = D >> 1;
    int Nvo = Do * Do * Do;
    int i = blockIdx.x * blockDim.x + threadIdx.x;
    if (i >= C * Nvo) return;
    int c = i / Nvo;
    int v = i - c * Nvo;
    int w  = v % Do;
    int hh = (v / Do) % Do;
    int d  = v / (Do * Do);
    const float* p = in + (size_t)c * D * D * D;
    float s = 0.f;
    for (int a = 0; a < 2; ++a)
        for (int b = 0; b < 2; ++b)
            for (int e = 0; e < 2; ++e)
                s += p[((size_t)(2 * d + a) * D + (2 * hh + b)) * D + (2 * w + e)];
    out[i] = s * 0.125f;
}

// ---------------- layernorm over channel, NCDHW -> channel-last ----------------
__global__ void ln_chlast_kernel(const float* __restrict__ in, const float* __restrict__ g,
                                 const float* __restrict__ b, float* __restrict__ out,
                                 int Nv, int C) {
    int v = blockIdx.x * blockDim.x + threadIdx.x;
    if (v >= Nv) return;
    float s = 0.f, s2 = 0.f;
    for (int c = 0; c < C; ++c) {
        float x = in[(size_t)c * Nv + v];
        s += x; s2 += x * x;
    }
    float m   = s / (float)C;
    float var = s2 / (float)C - m * m;
    float r   = rsqrtf(var + 1e-5f);
    for (int c = 0; c < C; ++c) {
        float x = in[(size_t)c * Nv + v];
        out[(size_t)v * C + c] = (x - m) * r * g[c] + b[c];
    }
}

// ---- cooperative weight staging into LDS (async-to-LDS on CDNA5 when available) ----
__device__ __forceinline__ void stage_weights(const float* __restrict__ W,
                                              const float* __restrict__ B,
                                              float* sW, float* sB, int wcount, int bcount) {
#ifdef USE_ASYNC_LDS
    for (int i = threadIdx.x; i < wcount; i += blockDim.x)
        __builtin_amdgcn_global_load_async_to_lds_b32((glb_i*)(W + i), (lds_i*)(sW + i), 0, 0);
    for (int i = threadIdx.x; i < bcount; i += blockDim.x)
        __builtin_amdgcn_global_load_async_to_lds_b32((glb_i*)(B + i), (lds_i*)(sB + i), 0, 0);
    __builtin_amdgcn_s_wait_asynccnt(0);
    __syncthreads();
#else
    for (int i = threadIdx.x; i < wcount; i += blockDim.x) sW[i] = W[i];
    for (int i = threadIdx.x; i < bcount; i += blockDim.x) sB[i] = B[i];
    __syncthreads();
#endif
}

// ---------------- WMMA f32 GEMM: O[M,N] = X[M,K] * W[K,N] + bias ----------------
// One wave32 computes a 16x16 tile via V_WMMA_F32_16X16X4_F32; W/bias staged in LDS.
__global__ void gemm_bias_wmma(const float* __restrict__ X, const float* __restrict__ W,
                               const float* __restrict__ B, float* __restrict__ O,
                               int M, int N, int K) {
    __shared__ float sW[32 * 96];
    __shared__ float sB[96];
    stage_weights(W, B, sW, sB, K * N, N);

    int gwave = (blockIdx.x * blockDim.x + threadIdx.x) >> 5;
    int lane  = threadIdx.x & 31;
    int Mt = M >> 4, Nt = N >> 4;
    if (gwave < Mt * Nt) {                                 // wave-uniform (after barrier)
        int tm = gwave / Nt, tn = gwave - tm * Nt;
        int hf = lane >> 4, l = lane & 15;
        const float* Xr = X + (size_t)(tm * 16 + l) * K;   // A: row = lane&15
        const float* Wc = sW + tn * 16 + l;                // B from LDS: col = lane&15
        v8f acc = {};
        for (int k0 = 0; k0 < K; k0 += 4) {
            int ka = k0 + 2 * hf;                          // lanes 0-15: K{0,1}; 16-31: K{2,3}
            v2f a, b;
            a.x = Xr[ka];         a.y = Xr[ka + 1];
            b.x = Wc[ka * N];     b.y = Wc[(ka + 1) * N];
            acc = __builtin_amdgcn_wmma_f32_16x16x4_f32(false, a, false, b, (short)0, acc,
                                                        false, false);
        }
        float bias = sB[tn * 16 + l];
        float* Or = O + (size_t)(tm * 16 + 8 * hf) * N + tn * 16 + l;
        #pragma unroll
        for (int r = 0; r < 8; ++r)
            Or[(size_t)r * N] = acc[r] + bias;             // D vgpr r -> M = r + 8*hf
    }
}

// ---------------- WMMA f32 proj GEMM with transpose-to-NCDHW + residual ----------------
__global__ void gemm_proj_wmma(const float* __restrict__ X, const float* __restrict__ W,
                               const float* __restrict__ B, const float* __restrict__ feat,
                               float* __restrict__ O, int M, int C) {
    __shared__ float sW[32 * 32];
    __shared__ float sB[32];
    stage_weights(W, B, sW, sB, C * C, C);

    int gwave = (blockIdx.x * blockDim.x + threadIdx.x) >> 5;
    int lane  = threadIdx.x & 31;
    int Mt = M >> 4, Nt = C >> 4;
    if (gwave < Mt * Nt) {
        int tm = gwave / Nt, tn = gwave - tm * Nt;
        int hf = lane >> 4, l = lane & 15;
        const float* Xr = X + (size_t)(tm * 16 + l) * C;
        const float* Wc = sW + tn * 16 + l;
        v8f acc = {};
        for (int k0 = 0; k0 < C; k0 += 4) {
            int ka = k0 + 2 * hf;
            v2f a, b;
            a.x = Xr[ka];         a.y = Xr[ka + 1];
            b.x = Wc[ka * C];     b.y = Wc[(ka + 1) * C];
            acc = __builtin_amdgcn_wmma_f32_16x16x4_f32(false, a, false, b, (short)0, acc,
                                                        false, false);
        }
        int n = tn * 16 + l;
        float bias = sB[n];
        #pragma unroll
        for (int r = 0; r < 8; ++r) {
            int m = tm * 16 + 8 * hf + r;
            O[(size_t)n * M + m] = acc[r] + bias + feat[(size_t)n * M + m];
        }
    }
}

// ---------------- 3x3x3 neighborhood attention, hd = 8 ----------------
__global__ void nat3d_attn_kernel(const float* __restrict__ qkv, float* __restrict__ out,
                                  int D, int C, int heads) {
    const int HD = 8;
    int Nv = D * D * D;
    int idx = blockIdx.x * blockDim.x + threadIdx.x;
    if (idx >= Nv * heads) return;
    int h = idx % heads;
    int v = idx / heads;
    int w = v % D;
    int y = (v / D) % D;
    int d = v / (D * D);
    const float scale = 0.35355339059327373f;   // 8^-0.5
    float q[HD];
    const float* qp = qkv + (size_t)v * 3 * C + h * HD;
    #pragma unroll
    for (int e = 0; e < HD; ++e) q[e] = qp[e] * scale;
    int sd = min(max(d - 1, 0), D - 3);
    int sh = min(max(y - 1, 0), D - 3);
    int sw = min(max(w - 1, 0), D - 3);
    float logit[27];
    int   nbr[27];
    float mx = -1e30f;
    int n = 0;
    for (int a = 0; a < 3; ++a)
        for (int b = 0; b < 3; ++b)
            for (int c2 = 0; c2 < 3; ++c2, ++n) {
                int nv = ((sd + a) * D + (sh + b)) * D + (sw + c2);
                nbr[n] = nv;
                const float* kp = qkv + (size_t)nv * 3 * C + C + h * HD;
                float dot = 0.f;
                #pragma unroll
                for (int e = 0; e < HD; ++e) dot += q[e] * kp[e];
                logit[n] = dot;
                mx = fmaxf(mx, dot);
            }
    float sum = 0.f;
    for (n = 0; n < 27; ++n) {
        float ee = expf(logit[n] - mx);
        logit[n] = ee;
        sum += ee;
    }
    float inv = 1.0f / sum;
    float o[HD] = {0, 0, 0, 0, 0, 0, 0, 0};
    for (n = 0; n < 27; ++n) {
        const float* vp = qkv + (size_t)nbr[n] * 3 * C + 2 * C + h * HD;
        float ww = logit[n] * inv;
        #pragma unroll
        for (int e = 0; e < HD; ++e) o[e] += ww * vp[e];
    }
    float* op = out + (size_t)v * C + h * HD;
    #pragma unroll
    for (int e = 0; e < HD; ++e) op[e] = o[e];
}

// ============================ host-side orchestration ============================

static void conv_block(const float* in, int Cin, int Cout, int D,
                       const float* w1, const float* b1, const float* w2, const float* b2,
                       float* t, float* y, float* out, float* stats, hipStream_t s) {
    int Nv = D * D * D;
    int total = Nv * Cout;
    int g = (total + 255) / 256;
    conv3d_kernel<<<g, 256, 0, s>>>(in, w1, b1, nullptr, t, Cin, Cout, D, 0);
    in_stats_kernel<<<Cout, 256, 0, s>>>(t, stats, Nv);
    norm_gelu_kernel<<<g, 256, 0, s>>>(t, stats, y, Nv, total);
    conv3d_kernel<<<g, 256, 0, s>>>(y, w2, b2, y, out, Cout, Cout, D, 1);  // fused gelu+res
}

static void na_block(float* cb, int C, int heads, int D,
                     const float* ln_g, const float* ln_b,
                     const float* wqkv, const float* bqkv,
                     const float* wproj, const float* bproj,
                     float* xl, float* qkv, float* att, float* out, hipStream_t s) {
    int Nv = D * D * D;
    ln_chlast_kernel<<<(Nv + 255) / 256, 256, 0, s>>>(cb, ln_g, ln_b, xl, Nv, C);
    int Mt = Nv >> 4;
    int waves = Mt * ((3 * C) >> 4);
    gemm_bias_wmma<<<(waves * 32 + 255) / 256, 256, 0, s>>>(xl, wqkv, bqkv, qkv, Nv, 3 * C, C);
    nat3d_attn_kernel<<<(Nv * heads + 255) / 256, 256, 0, s>>>(qkv, att, D, C, heads);
    waves = Mt * (C >> 4);
    gemm_proj_wmma<<<(waves * 32 + 255) / 256, 256, 0, s>>>(att, wproj, bproj, cb, out, Nv, C);
}

extern "C" void kernel_launch(void* const* d_in, const int* in_sizes, int n_in,
                              void* d_out, int out_size, void* d_ws, size_t ws_size,
                              hipStream_t stream) {
    (void)in_sizes; (void)n_in; (void)out_size; (void)ws_size;
    const float* x = (const float*)d_in[0];
    // conv0
    const float *c0w1 = (const float*)d_in[1],  *c0b1 = (const float*)d_in[2];
    const float *c0w2 = (const float*)d_in[3],  *c0b2 = (const float*)d_in[4];
    // conv1
    const float *c1w1 = (const float*)d_in[5],  *c1b1 = (const float*)d_in[6];
    const float *c1w2 = (const float*)d_in[7],  *c1b2 = (const float*)d_in[8];
    // na1
    const float *n1g  = (const float*)d_in[9],  *n1b  = (const float*)d_in[10];
    const float *n1wq = (const float*)d_in[11], *n1bq = (const float*)d_in[12];
    const float *n1wp = (const float*)d_in[13], *n1bp = (const float*)d_in[14];
    // conv2
    const float *c2w1 = (const float*)d_in[15], *c2b1 = (const float*)d_in[16];
    const float *c2w2 = (const float*)d_in[17], *c2b2 = (const float*)d_in[18];
    // na2
    const float *n2g  = (const float*)d_in[19], *n2b  = (const float*)d_in[20];
    const float *n2wq = (const float*)d_in[21], *n2bq = (const float*)d_in[22];
    const float *n2wp = (const float*)d_in[23], *n2bp = (const float*)d_in[24];
    // conv3
    const float *c3w1 = (const float*)d_in[25], *c3b1 = (const float*)d_in[26];
    const float *c3w2 = (const float*)d_in[27], *c3b2 = (const float*)d_in[28];
    // na3
    const float *n3g  = (const float*)d_in[29], *n3b  = (const float*)d_in[30];
    const float *n3wq = (const float*)d_in[31], *n3bq = (const float*)d_in[32];
    const float *n3wp = (const float*)d_in[33], *n3bp = (const float*)d_in[34];

    // workspace carve-out (floats)
    float* ws    = (float*)d_ws;
    float* t     = ws;                       // 7,077,888
    float* y     = ws + 7077888;             // 7,077,888
    float* h     = ws + 14155776;            //   884,736 (pooled)
    float* cb    = ws + 15040512;            // 1,769,472 (conv-block output / NA residual)
    float* stats = ws + 16809984;            //        64

    float* out  = (float*)d_out;
    float* out0 = out;                       // 8  x 96^3
    float* out1 = out0 + 7077888;            // 16 x 48^3
    float* out2 = out1 + 1769472;            // 32 x 24^3
    float* out3 = out2 + 442368;             // 32 x 12^3

    float* xl  = t;                          // channel-last LN output (<= 1,769,472)
    float* att = t + 2097152;                // attention output       (<= 1,769,472)
    float* qkv = y;                          // qkv buffer             (<= 5,308,416)

    // ---- stage 0: conv0 block @ 96^3 (1 -> 8) ----
    conv_block(x, 1, 8, 96, c0w1, c0b1, c0w2, c0b2, t, y, out0, stats, stream);
    avgpool2_kernel<<<(8 * 48 * 48 * 48 + 255) / 256, 256, 0, stream>>>(out0, h, 8, 96);

    // ---- stage 1: conv1 block @ 48^3 (8 -> 16), NA heads=2 ----
    conv_block(h, 8, 16, 48, c1w1, c1b1, c1w2, c1b2, t, y, cb, stats, stream);
    na_block(cb, 16, 2, 48, n1g, n1b, n1wq, n1bq, n1wp, n1bp, xl, qkv, att, out1, stream);
    avgpool2_kernel<<<(16 * 24 * 24 * 24 + 255) / 256, 256, 0, stream>>>(out1, h, 16, 48);

    // ---- stage 2: conv2 block @ 24^3 (16 -> 32), NA heads=4 ----
    conv_block(h, 16, 32, 24, c2w1, c2b1, c2w2, c2b2, t, y, cb, stats, stream);
    na_block(cb, 32, 4, 24, n2g, n2b, n2wq, n2bq, n2wp, n2bp, xl, qkv, att, out2, stream);
    avgpool2_kernel<<<(32 * 12 * 12 * 12 + 255) / 256, 256, 0, stream>>>(out2, h, 32, 24);

    // ---- stage 3: conv3 block @ 12^3 (32 -> 32), NA heads=4 ----
    conv_block(h, 32, 32, 12, c3w1, c3b1, c3w2, c3b2, t, y, cb, stats, stream);
    na_block(cb, 32, 4, 12, n3g, n3b, n3wq, n3bq, n3wp, n3bp, xl, qkv, att, out3, stream);
}